// FFM_21912923144768
// MI455X (gfx1250) — compile-verified
//
#include <hip/hip_runtime.h>
#include <hip/hip_bf16.h>

// ---------------- problem constants (from reference) ----------------
#define DENSE_N   13
#define SPARSE_N  26
#define EMB_DIM   16
#define VOCAB     100000
#define BATCH     4096
#define FEAT_N    (DENSE_N + SPARSE_N * EMB_DIM)   // 429
#define FIELD_N   (DENSE_N + SPARSE_N)             // 39
#define P         448                              // padded dim: 28 tiles of 16

typedef __attribute__((ext_vector_type(2))) float v2f;
typedef __attribute__((ext_vector_type(8))) float v8f;

// field id exactly as the reference: f = n<14 ? n : (n-14)/16 + 14, clipped to 38
__device__ __forceinline__ int field_id(int n) {
    int f = (n < 14) ? n : (((n - 14) >> 4) + 14);
    return (f > FIELD_N - 1) ? (FIELD_N - 1) : f;
}

// ---------------- kernel 1: gather features into padded F (BATCH x P) ----------------
__global__ __launch_bounds__(256) void ffm_build_feat(
    const float* __restrict__ dense, const int* __restrict__ ids,
    const float* __restrict__ emb, float* __restrict__ F) {
    int idx = blockIdx.x * blockDim.x + threadIdx.x;
    if (idx >= BATCH * P) return;
    int b = idx / P, j = idx - b * P;
    float v = 0.0f;
    if (j < DENSE_N) {
        v = dense[b * DENSE_N + j];
    } else if (j < FEAT_N) {
        int e = j - DENSE_N;
        int s = e >> 4, d = e & 15;
        int row = ids[b * SPARSE_N + s];
        v = emb[((size_t)s * VOCAB + (size_t)row) * EMB_DIM + d];
    }
    F[(size_t)b * P + j] = v;
}

// ---------------- kernel 2: build masked interaction matrix, stored TRANSPOSED ----------
// Mt[j*P + i] = (i>=1 && j>i, j<429) ? sum_d fm[i,f(j),d]*fm[j,f(i),d] : 0
__global__ __launch_bounds__(256) void ffm_build_m(
    const float* __restrict__ fm, float* __restrict__ Mt) {
    int idx = blockIdx.x * blockDim.x + threadIdx.x;
    if (idx >= P * P) return;
    int i = idx / P, j = idx - i * P;
    float s = 0.0f;
    if (i >= 1 && j > i && j < FEAT_N) {
        const float* a = fm + ((size_t)i * FIELD_N + field_id(j)) * EMB_DIM;
        const float* b = fm + ((size_t)j * FIELD_N + field_id(i)) * EMB_DIM;
        #pragma unroll
        for (int d = 0; d < EMB_DIM; ++d) s = fmaf(a[d], b[d], s);
    }
    Mt[(size_t)j * P + i] = s;
}

// ---------------- kernel 3: V = F @ M via V_WMMA_F32_16X16X4_F32 ----------------
// Each wave: one 16-row x 64-col output slab. A-frag reused across 4 column tiles.
// A layout (16x4 f32): lane r=lane&15 selects row; half=lane>>4 selects K pair {0,1}/{2,3}
//   -> one contiguous b64 load per lane: F[row0+r][k + 2*half .. +1]
// B layout (4x16 f32, mirrored): lanes 0-15 K={0,1}, lanes 16-31 K={2,3}, N=lane&15
//   -> with M stored transposed: Mt[(col)*P + k + 2*half .. +1], contiguous b64 load.
__global__ __launch_bounds__(256) void ffm_gemm(
    const float* __restrict__ F, const float* __restrict__ Mt,
    float* __restrict__ V) {
    const int wave = blockIdx.x * (blockDim.x >> 5) + (threadIdx.x >> 5);
    const int lane = threadIdx.x & 31;
    const int row_block = wave / 7;              // 256 row blocks of 16
    const int col_group = wave - row_block * 7;  // 7 groups of 64 cols
    const int row0 = row_block * 16;
    const int c0   = col_group * 64;
    const int r    = lane & 15;
    const int half = lane >> 4;

    const float* a_ptr = F  + (size_t)(row0 + r) * P + 2 * half;
    const float* b_ptr = Mt + (size_t)(c0 + r)   * P + 2 * half;

    v8f acc0 = {}, acc1 = {}, acc2 = {}, acc3 = {};

    #pragma unroll 4
    for (int k = 0; k < P; k += 4) {
        // WGP-scope prefetch (near cache), one cacheline (16 floats), 2 lines ahead,
        // issued once per 16-float group (condition is compile-time under unroll-4).
        if ((k & 15) == 0) {
            __builtin_prefetch(a_ptr + k + 32, 0, 3);
            __builtin_prefetch(b_ptr + k + 32, 0, 3);
        }
        v2f a  = *(const v2f*)(a_ptr + k);
        v2f b0 = *(const v2f*)(b_ptr + (size_t)(0 * 16) * P + k);
        v2f b1 = *(const v2f*)(b_ptr + (size_t)(1 * 16) * P + k);
        v2f b2 = *(const v2f*)(b_ptr + (size_t)(2 * 16) * P + k);
        v2f b3 = *(const v2f*)(b_ptr + (size_t)(3 * 16) * P + k);
        acc0 = __builtin_amdgcn_wmma_f32_16x16x4_f32(false, a, false, b0, (short)0, acc0, false, false);
        acc1 = __builtin_amdgcn_wmma_f32_16x16x4_f32(false, a, false, b1, (short)0, acc1, false, false);
        acc2 = __builtin_amdgcn_wmma_f32_16x16x4_f32(false, a, false, b2, (short)0, acc2, false, false);
        acc3 = __builtin_amdgcn_wmma_f32_16x16x4_f32(false, a, false, b3, (short)0, acc3, false, false);
    }

    // D layout: VGPR v, lane<16 -> (row0+v, c+lane); lane>=16 -> (row0+v+8, c+lane-16)
    float* vout = V + (size_t)(row0 + 8 * half) * P + (c0 + r);
    #pragma unroll
    for (int v = 0; v < 8; ++v) {
        vout[(size_t)v * P +  0] = acc0[v];
        vout[(size_t)v * P + 16] = acc1[v];
        vout[(size_t)v * P + 32] = acc2[v];
        vout[(size_t)v * P + 48] = acc3[v];
    }
}

// ---------------- kernel 4: out[b] = sigmoid(bias + sum_j F[b,j]*(V[b,j]+w[j])) -------
__global__ __launch_bounds__(256) void ffm_reduce(
    const float* __restrict__ F, const float* __restrict__ V,
    const float* __restrict__ lin_w, const float* __restrict__ lin_b,
    float* __restrict__ out) {
    int b    = blockIdx.x * (blockDim.x >> 5) + (threadIdx.x >> 5); // one wave per row
    int lane = threadIdx.x & 31;
    if (b >= BATCH) return;
    const float* f = F + (size_t)b * P;
    const float* v = V + (size_t)b * P;
    float s = 0.0f;
    #pragma unroll
    for (int j0 = 0; j0 < P; j0 += 32) {
        int j = j0 + lane;
        float wj = (j < FEAT_N) ? lin_w[j] : 0.0f;
        s = fmaf(f[j], v[j] + wj, s);
    }
    #pragma unroll
    for (int off = 16; off > 0; off >>= 1) s += __shfl_xor(s, off, 32);
    if (lane == 0) out[b] = 1.0f / (1.0f + __expf(-(s + lin_b[0])));
}

// ---------------- host launcher ----------------
extern "C" void kernel_launch(void* const* d_in, const int* in_sizes, int n_in,
                              void* d_out, int out_size, void* d_ws, size_t ws_size,
                              hipStream_t stream) {
    const float* dense  = (const float*)d_in[0];  // (4096,13)
    const int*   ids    = (const int*)  d_in[1];  // (4096,26)
    const float* emb    = (const float*)d_in[2];  // (26,100000,16)
    const float* fm     = (const float*)d_in[3];  // (429,39,16)
    const float* lin_w  = (const float*)d_in[4];  // (429,)
    const float* lin_b  = (const float*)d_in[5];  // (1,)
    float*       out    = (float*)d_out;          // (4096,)

    // workspace layout (floats): F | Mt | V  (~15.5 MB total)
    float* F  = (float*)d_ws;
    float* Mt = F  + (size_t)BATCH * P;
    float* V  = Mt + (size_t)P * P;

    {   // gather features into padded F
        int n = BATCH * P;
        ffm_build_feat<<<(n + 255) / 256, 256, 0, stream>>>(dense, ids, emb, F);
    }
    {   // masked interaction matrix (transposed for contiguous B-fragments)
        int n = P * P;
        ffm_build_m<<<(n + 255) / 256, 256, 0, stream>>>(fm, Mt);
    }
    {   // V = F @ M via fp32 WMMA: 256 row-blocks * 7 col-groups = 1792 waves, 8/WG
        ffm_gemm<<<(256 * 7) / 8, 256, 0, stream>>>(F, Mt, V);
    }
    {   // per-row quadratic-form contraction + linear + sigmoid
        ffm_reduce<<<(BATCH + 7) / 8, 256, 0, stream>>>(F, V, lin_w, lin_b, out);
    }
}